// DCNv3_pytorch_12008728560141
// MI455X (gfx1250) — compile-verified
//
#include <hip/hip_runtime.h>
#include <math.h>

// Problem constants (match reference config)
#define NB   8
#define HH   64
#define WW   64
#define CC   256
#define GG   16
#define GCC  16
#define PP   9
#define NPIX (NB * HH * WW)      // 32768
#define KK   256                 // K dim of every GEMM in this model

typedef __attribute__((ext_vector_type(2))) float v2f;
typedef __attribute__((ext_vector_type(4))) float v4f;
typedef __attribute__((ext_vector_type(8))) float v8f;

// ---------------------------------------------------------------------------
// GEMM on the matrix pipe (full fp32 WMMA, 16x16x4):
//   C[M x Nout] = A[M x KK] @ W^T + bias,  W stored row-major (Nout x KK).
// One wave -> 16 rows x NT*16 cols. Block = 8 waves = 128 rows.
// K-loop: groups of GS=4 k-steps (20 b64 loads per group, burst-issued),
// double-buffered so group i+1's loads fly while group i's 16 WMMAs drain.
// Fragment layouts per CDNA5 ISA (wave32):
//   A 16x4 : lane l -> row l&15, K-pair (l>>4)*2          (v2f, contiguous)
//   B 4x16 : lane l -> col l&15, K-pair (l>>4)*2          (v2f, contiguous in W^T)
//   C 16x16: lane l, VGPR r -> row (l>>4)*8 + r, col l&15
// ---------------------------------------------------------------------------
#define GS 4                      // k-steps per group (16 K values)
#define NGRP (KK / (4 * GS))      // 16 groups

template<int NT>
__device__ __forceinline__ void load_group(const float* __restrict__ arow,
                                           const float* const (&brow)[NT], int kbase,
                                           v2f (&ag)[GS], v2f (&bg)[GS][NT])
{
    #pragma unroll
    for (int s = 0; s < GS; ++s) {
        ag[s] = *(const v2f*)(arow + kbase + 4 * s);
        #pragma unroll
        for (int t = 0; t < NT; ++t)
            bg[s][t] = *(const v2f*)(brow[t] + kbase + 4 * s);
    }
}

template<int NT>
__device__ __forceinline__ void compute_group(v2f (&ag)[GS], v2f (&bg)[GS][NT], v8f (&acc)[NT])
{
    #pragma unroll
    for (int s = 0; s < GS; ++s) {
        #pragma unroll
        for (int t = 0; t < NT; ++t)
            acc[t] = __builtin_amdgcn_wmma_f32_16x16x4_f32(
                false, ag[s], false, bg[s][t], (short)0, acc[t], false, false);
    }
}

template<int NT>
__global__ __launch_bounds__(256)
void gemm_wmma_f32(const float* __restrict__ A, const float* __restrict__ Wt,
                   const float* __restrict__ bias, float* __restrict__ Cout,
                   int Nout, int n_base)
{
    const int lane = threadIdx.x & 31;
    const int wave = threadIdx.x >> 5;
    const int m0   = (blockIdx.y * 8 + wave) * 16;
    const int n0   = n_base + blockIdx.x * (NT * 16);

    const int mrow  = lane & 15;
    const int khalf = (lane >> 4) << 1;          // 0 or 2
    const int ncol  = lane & 15;

    const float* arow = A + (size_t)(m0 + mrow) * KK + khalf;
    const float* brow[NT];
    #pragma unroll
    for (int t = 0; t < NT; ++t)
        brow[t] = Wt + (size_t)(n0 + t * 16 + ncol) * KK + khalf;

    v8f acc[NT] = {};

    v2f aA[GS], aB[GS];
    v2f bA[GS][NT], bB[GS][NT];

    // prologue: group 0 in flight
    load_group<NT>(arow, brow, 0, aA, bA);

    // ping-pong over groups 1..14 (7 pairs)
    #pragma unroll 1
    for (int g = 0; g < (NGRP - 2) / 2; ++g) {
        load_group<NT>(arow, brow, (2 * g + 1) * 4 * GS, aB, bB);
        compute_group<NT>(aA, bA, acc);
        load_group<NT>(arow, brow, (2 * g + 2) * 4 * GS, aA, bA);
        compute_group<NT>(aB, bB, acc);
    }
    // epilogue: group 15
    load_group<NT>(arow, brow, (NGRP - 1) * 4 * GS, aB, bB);
    compute_group<NT>(aA, bA, acc);
    compute_group<NT>(aB, bB, acc);

    const int mbase = m0 + (lane >> 4) * 8;
    #pragma unroll
    for (int t = 0; t < NT; ++t) {
        const int col = n0 + t * 16 + ncol;
        const float bv = bias[col];
        #pragma unroll
        for (int r = 0; r < 8; ++r) {
            Cout[(size_t)(mbase + r) * Nout + col] = acc[t][r] + bv;
        }
    }
}

// ---------------------------------------------------------------------------
// Depthwise 3x3 conv (zero pad) + LayerNorm over C + exact GELU.
// One 256-thread block per pixel; thread = channel. LDS tree reduction.
// ---------------------------------------------------------------------------
__global__ __launch_bounds__(256)
void dwconv_ln_gelu(const float* __restrict__ x,  const float* __restrict__ dww,
                    const float* __restrict__ dwb, const float* __restrict__ lng,
                    const float* __restrict__ lnb, float* __restrict__ x1)
{
    const int pix = blockIdx.x;
    const int c   = threadIdx.x;
    const int w   = pix & (WW - 1);
    const int h   = (pix >> 6) & (HH - 1);
    const int n   = pix >> 12;

    float acc = dwb[c];
    #pragma unroll
    for (int i = 0; i < 3; ++i) {
        const int hh = h + i - 1;
        if ((unsigned)hh < (unsigned)HH) {
            #pragma unroll
            for (int j = 0; j < 3; ++j) {
                const int ww = w + j - 1;
                if ((unsigned)ww < (unsigned)WW) {
                    acc += x[(((size_t)(n * HH + hh)) * WW + ww) * CC + c]
                           * dww[c * 9 + i * 3 + j];
                }
            }
        }
    }

    __shared__ float s_sum[256];
    __shared__ float s_sq[256];
    s_sum[c] = acc;
    s_sq[c]  = acc * acc;
    __syncthreads();
    #pragma unroll
    for (int off = 128; off > 0; off >>= 1) {
        if (c < off) { s_sum[c] += s_sum[c + off]; s_sq[c] += s_sq[c + off]; }
        __syncthreads();
    }
    const float mean = s_sum[0] * (1.0f / CC);
    const float var  = s_sq[0] * (1.0f / CC) - mean * mean;
    const float yv   = (acc - mean) * rsqrtf(var + 1e-6f) * lng[c] + lnb[c];
    const float gel  = 0.5f * yv * (1.0f + erff(yv * 0.70710678118654752f));
    x1[(size_t)pix * CC + c] = gel;
}

// ---------------------------------------------------------------------------
// Softmax over P=9 taps for each (pixel, group). Logits are contiguous.
// ---------------------------------------------------------------------------
__global__ __launch_bounds__(256)
void mask_softmax(float* __restrict__ m)
{
    const int idx = blockIdx.x * 256 + threadIdx.x;     // pix*G + g
    if (idx >= NPIX * GG) return;
    float* p = m + (size_t)idx * PP;
    float v[PP];
    float mx = -3.402823e38f;
    #pragma unroll
    for (int i = 0; i < PP; ++i) { v[i] = p[i]; mx = fmaxf(mx, v[i]); }
    float s = 0.f;
    #pragma unroll
    for (int i = 0; i < PP; ++i) { v[i] = expf(v[i] - mx); s += v[i]; }
    const float inv = 1.0f / s;
    #pragma unroll
    for (int i = 0; i < PP; ++i) p[i] = v[i] * inv;
}

// ---------------------------------------------------------------------------
// DCNv3 core, 4 channels per thread (float4 gathers, b128 traffic).
// In padded coordinates the reference sampling collapses to
//   gx = w + p/3 + off_x,  gy = h + p%3 + off_y,
// and a corner contributes iff it lands in the interior [1,64]x[1,64]
// (zero padding and the validity mask coincide).
// ---------------------------------------------------------------------------
__device__ __forceinline__ v4f samp4(const float* __restrict__ base, int xi, int yi)
{
    if ((unsigned)(xi - 1) < (unsigned)WW && (unsigned)(yi - 1) < (unsigned)HH)
        return *(const v4f*)(base + ((size_t)(yi - 1) * WW + (xi - 1)) * CC);
    return (v4f){};
}

__global__ __launch_bounds__(256)
void dcn_core(const float* __restrict__ xp, const float* __restrict__ off,
              const float* __restrict__ msk, float* __restrict__ core)
{
    const int idx = blockIdx.x * 256 + threadIdx.x;   // pix*(C/4) + c4
    const int c4  = idx & (CC / 4 - 1);               // 0..63
    const int pix = idx >> 6;
    const int c   = c4 << 2;
    const int w   = pix & (WW - 1);
    const int h   = (pix >> 6) & (HH - 1);
    const int n   = pix >> 12;
    const int g   = c >> 4;

    const float* op   = off + (size_t)pix * (GG * PP * 2) + g * (PP * 2);
    const float* mp   = msk + (size_t)pix * (GG * PP) + g * PP;
    const float* base = xp + (size_t)n * HH * WW * CC + c;

    v4f acc = {};
    #pragma unroll
    for (int p = 0; p < PP; ++p) {
        const float gx = (float)(w + (p / 3)) + op[2 * p];
        const float gy = (float)(h + (p % 3)) + op[2 * p + 1];
        const float x0f = floorf(gx), y0f = floorf(gy);
        const float tx = gx - x0f, ty = gy - y0f;
        const int x0 = (int)x0f, y0 = (int)y0f;
        const v4f v00 = samp4(base, x0,     y0);
        const v4f v10 = samp4(base, x0 + 1, y0);
        const v4f v01 = samp4(base, x0,     y0 + 1);
        const v4f v11 = samp4(base, x0 + 1, y0 + 1);
        const float w00 = (1.f - tx) * (1.f - ty);
        const float w10 = tx * (1.f - ty);
        const float w01 = (1.f - tx) * ty;
        const float w11 = tx * ty;
        const v4f bil = v00 * w00 + v10 * w10 + v01 * w01 + v11 * w11;
        acc += mp[p] * bil;
    }
    *(v4f*)(core + (size_t)pix * CC + c) = acc;
}

// ---------------------------------------------------------------------------
extern "C" void kernel_launch(void* const* d_in, const int* in_sizes, int n_in,
                              void* d_out, int out_size, void* d_ws, size_t ws_size,
                              hipStream_t stream)
{
    const float* x      = (const float*)d_in[0];
    const float* dw_w   = (const float*)d_in[1];
    const float* dw_b   = (const float*)d_in[2];
    const float* ln_g   = (const float*)d_in[3];
    const float* ln_b   = (const float*)d_in[4];
    const float* off_w  = (const float*)d_in[5];
    const float* off_b  = (const float*)d_in[6];
    const float* mask_w = (const float*)d_in[7];
    const float* mask_b = (const float*)d_in[8];
    const float* in_w   = (const float*)d_in[9];
    const float* in_b   = (const float*)d_in[10];
    const float* out_w  = (const float*)d_in[11];
    const float* out_b  = (const float*)d_in[12];
    float* out = (float*)d_out;

    // Workspace layout (fp32):
    float* ws   = (float*)d_ws;
    float* xp   = ws;                                   // NPIX*256  values (input_proj)
    float* x1   = xp   + (size_t)NPIX * CC;             // NPIX*256  dw+LN+GELU
    float* offb = x1   + (size_t)NPIX * CC;             // NPIX*288  offsets
    float* mkb  = offb + (size_t)NPIX * (GG * PP * 2);  // NPIX*144  mask (softmax in-place)
    float* core = mkb  + (size_t)NPIX * (GG * PP);      // NPIX*256  dcn output

    const dim3 blk(256);
    const int mgrid = NPIX / 128;                       // 8 waves x 16 rows per block

    // 1) value projection: xp = x @ in_w^T + in_b  (N=256 = 4 x 64)
    gemm_wmma_f32<4><<<dim3(4, mgrid), blk, 0, stream>>>(x, in_w, in_b, xp, 256, 0);
    // 2) dw conv + LN + GELU -> x1
    dwconv_ln_gelu<<<dim3(NPIX), blk, 0, stream>>>(x, dw_w, dw_b, ln_g, ln_b, x1);
    // 3) offset head: N=288 = 4 x 64 + 1 x 32
    gemm_wmma_f32<4><<<dim3(4, mgrid), blk, 0, stream>>>(x1, off_w, off_b, offb, 288, 0);
    gemm_wmma_f32<2><<<dim3(1, mgrid), blk, 0, stream>>>(x1, off_w, off_b, offb, 288, 256);
    // 4) mask head: N=144 = 2 x 64 + 1 x 16
    gemm_wmma_f32<4><<<dim3(2, mgrid), blk, 0, stream>>>(x1, mask_w, mask_b, mkb, 144, 0);
    gemm_wmma_f32<1><<<dim3(1, mgrid), blk, 0, stream>>>(x1, mask_w, mask_b, mkb, 144, 128);
    // 5) softmax over P per (pixel, group)
    mask_softmax<<<dim3((NPIX * GG) / 256), blk, 0, stream>>>(mkb);
    // 6) deformable sampling core (4 channels / thread)
    dcn_core<<<dim3((NPIX * CC / 4) / 256), blk, 0, stream>>>(xp, offb, mkb, core);
    // 7) output projection straight into d_out
    gemm_wmma_f32<4><<<dim3(4, mgrid), blk, 0, stream>>>(core, out_w, out_b, out, 256, 0);
}